// partial_conv_64896955842689
// MI455X (gfx1250) — compile-verified
//
#include <hip/hip_runtime.h>

typedef __attribute__((ext_vector_type(2))) float v2f;
typedef __attribute__((ext_vector_type(8))) float v8f;
typedef __attribute__((__vector_size__(16))) int v4i_;

#define H_SZ   384
#define W_PIX  384
#define CH     64
#define OC     64
#define KTOT   576                  // CH * 9
#define HWSZ   (H_SZ * W_PIX)
#define TILE_W 32
#define ROWS_B 4                    // output rows per block
#define XROWS  6                    // input halo rows (ROWS_B + 2)
#define XCOLS  36                   // 34 halo cols padded to 36 (bank spread)
#define SW_ELEMS (OC * KTOT)        // 36864 floats = 147456 B
#define SX_ELEMS (XROWS * CH * XCOLS) // 13824 floats = 55296 B
#define SV_ELEMS (ROWS_B * TILE_W)  // 128 validity flags

#if defined(__has_builtin)
#if __has_builtin(__builtin_amdgcn_global_load_async_to_lds_b128) && \
    __has_builtin(__builtin_amdgcn_s_wait_asynccnt)
#define HAVE_ASYNC_LDS 1
#endif
#endif

// Pre-pass: reorder OIHW weights -> K-major [oc][kh][kw][ic] so the main
// kernel's LDS staging is a pure contiguous 128b copy.
__global__ void reorder_weights_kernel(const float* __restrict__ W,
                                       float* __restrict__ Wk) {
    int t = blockIdx.x * blockDim.x + threadIdx.x;
    if (t >= SW_ELEMS) return;
    int oc = t / KTOT;
    int r  = t % KTOT;
    int kh = r / 192;
    int kw = (r / 64) % 3;
    int ic = r % 64;
    Wk[t] = W[oc * (CH * 9) + ic * 9 + kh * 3 + kw];
}

__global__ __launch_bounds__(256)
void partial_conv_wmma_kernel(const float* __restrict__ x,
                              const float* __restrict__ m,
                              const float* __restrict__ Wk,
                              const float* __restrict__ bias,
                              float* __restrict__ out) {
    extern __shared__ float smem[];
    float* sW = smem;                  // [OC][KTOT]          K-major weights
    float* sX = smem + SW_ELEMS;       // [XROWS][CH][XCOLS]  halo tile
    float* sV = sX + SX_ELEMS;         // [ROWS_B][TILE_W]    per-pixel validity

    const int w0  = blockIdx.x * TILE_W;
    const int h0  = blockIdx.y * ROWS_B;
    const int tid = threadIdx.x;

    // ---- stage weights (147 KB contiguous) ----
#if HAVE_ASYNC_LDS
    {
        const float4* src = (const float4*)Wk;
        float4* dst = (float4*)sW;
        for (int i = tid; i < SW_ELEMS / 4; i += 256) {
            __builtin_amdgcn_global_load_async_to_lds_b128(
                (v4i_*)(src + i), (v4i_*)(dst + i), 0, 0);
        }
    }
#else
    {
        const float4* src = (const float4*)Wk;
        float4* dst = (float4*)sW;
        #pragma unroll 4
        for (int i = tid; i < SW_ELEMS / 4; i += 256) dst[i] = src[i];
    }
#endif

    // ---- stage x halo tile [6][64][34] (rows h0-1..h0+4, cols w0-1..w0+32) ----
    for (int i = tid; i < XROWS * CH * 34; i += 256) {
        int dr  = i / (CH * 34);
        int rem = i % (CH * 34);
        int ic  = rem / 34;
        int col = rem % 34;
        int gh = h0 + dr - 1;
        int gw = w0 + col - 1;
        float v = 0.0f;
        if (gh >= 0 && gh < H_SZ && gw >= 0 && gw < W_PIX)
            v = x[ic * HWSZ + gh * W_PIX + gw];
        sX[dr * (CH * XCOLS) + ic * XCOLS + col] = v;
    }

    // ---- per-pixel mask validity: 3x3 window max of m > 0 ----
    if (tid < SV_ELEMS) {
        int rr = tid >> 5;
        int cc = tid & 31;
        int ph = h0 + rr;
        int pw = w0 + cc;
        float mx = 0.0f;
        #pragma unroll
        for (int dh = -1; dh <= 1; ++dh) {
            int gh = ph + dh;
            if (gh < 0 || gh >= H_SZ) continue;
            #pragma unroll
            for (int dw = -1; dw <= 1; ++dw) {
                int gw = pw + dw;
                if (gw < 0 || gw >= W_PIX) continue;
                mx = fmaxf(mx, m[gh * W_PIX + gw]);
            }
        }
        sV[tid] = mx;
    }
#if HAVE_ASYNC_LDS
    __builtin_amdgcn_s_wait_asynccnt(0);
#endif
    __syncthreads();

    // ---- wave tiling: 4 oc-tiles x 2 wave-halves; each wave: 4 pixel tiles ----
    // pixel tile index px (0..7): row rr = px>>1, col base c0 = (px&1)*16
    const int wave = tid >> 5;
    const int lane = tid & 31;
    const int g    = lane >> 4;      // lane group (0: lanes 0-15, 1: 16-31)
    const int ln   = lane & 15;
    const int oc0  = (wave & 3) * 16;
    const int half = wave >> 2;      // handles px tiles half*4 .. half*4+3

    v8f acc[4] = {};

    // A fragment (16x4 f32): lane ln = row M=oc0+ln; VGPR v holds K=2g+v
    //   -> one contiguous float2 read, reused by all 4 accumulators.
    // B fragment (4x16 f32): VGPR v, group g holds K=kbase+2g+v at column n.
    const float* wrow = &sW[(oc0 + ln) * KTOT + 2 * g];

    for (int kh = 0; kh < 3; ++kh) {
        for (int kw = 0; kw < 3; ++kw) {
            const float* wk = wrow + kh * 192 + kw * 64;
            const float* xc[4];
            #pragma unroll
            for (int j = 0; j < 4; ++j) {
                int px = half * 4 + j;
                int rr = px >> 1;
                int c0 = (px & 1) * 16;
                xc[j] = &sX[(rr + kh) * (CH * XCOLS) + (2 * g) * XCOLS
                            + (c0 + ln + kw)];
            }
            #pragma unroll 4
            for (int ic = 0; ic < CH; ic += 4) {
                v2f a = *(const v2f*)(wk + ic);
                #pragma unroll
                for (int j = 0; j < 4; ++j) {
                    v2f b;
                    b.x = xc[j][ic * XCOLS];          // channel ic + 2g
                    b.y = xc[j][ic * XCOLS + XCOLS];  // channel ic + 2g + 1
                    acc[j] = __builtin_amdgcn_wmma_f32_16x16x4_f32(
                        false, a, false, b, (short)0, acc[j], false, false);
                }
            }
        }
    }

    // ---- epilogue: D layout VGPR r -> M = r + 8*g, N = ln ----
    float bv[8];
    #pragma unroll
    for (int r = 0; r < 8; ++r) bv[r] = bias[oc0 + r + 8 * g];

    #pragma unroll
    for (int j = 0; j < 4; ++j) {
        int px = half * 4 + j;
        int rr = px >> 1;
        int c0 = (px & 1) * 16;
        int gh = h0 + rr;
        int gw = w0 + c0 + ln;
        bool valid = sV[rr * TILE_W + c0 + ln] > 0.0f;
        #pragma unroll
        for (int r = 0; r < 8; ++r) {
            int oc = oc0 + r + 8 * g;
            size_t idx = (size_t)oc * HWSZ + (size_t)gh * W_PIX + gw;
            float conv = acc[j][r] + bv[r];
            float base = x[idx];
            out[idx] = valid ? conv : base;
        }
    }
}

extern "C" void kernel_launch(void* const* d_in, const int* in_sizes, int n_in,
                              void* d_out, int out_size, void* d_ws, size_t ws_size,
                              hipStream_t stream) {
    (void)in_sizes; (void)n_in; (void)out_size; (void)ws_size;
    const float* x    = (const float*)d_in[0];
    const float* m    = (const float*)d_in[1];
    const float* W    = (const float*)d_in[2];
    const float* bias = (const float*)d_in[3];
    float* out = (float*)d_out;
    float* Wk  = (float*)d_ws;   // 147456 B of K-major weights

    reorder_weights_kernel<<<(SW_ELEMS + 255) / 256, 256, 0, stream>>>(W, Wk);

    size_t smem_bytes = (size_t)(SW_ELEMS + SX_ELEMS + SV_ELEMS) * sizeof(float);
    (void)hipFuncSetAttribute((const void*)partial_conv_wmma_kernel,
                              hipFuncAttributeMaxDynamicSharedMemorySize,
                              (int)smem_bytes);

    dim3 grid(W_PIX / TILE_W, H_SZ / ROWS_B);
    partial_conv_wmma_kernel<<<grid, 256, smem_bytes, stream>>>(x, m, Wk, bias, out);
}